// DeformAlign_4466765988319
// MI455X (gfx1250) — compile-verified
//
#include <hip/hip_runtime.h>

#define W_   96
#define H_   96
#define HW   9216        // 96*96
#define CC   64
#define KT   9
#define BB   4
#define QDIM 576         // CC*KT  (GEMM inner dim)

typedef __attribute__((ext_vector_type(2))) float v2f;
typedef __attribute__((ext_vector_type(8))) float v8f;

// ---------------------------------------------------------------------------
// Kernel 1: offset conv  (3x3, 64 -> 18 channels, zero padding)
// out[p] laid out as [B][18][H][W], p covers B*18*HW = 663552 threads exactly.
// ---------------------------------------------------------------------------
__global__ __launch_bounds__(256) void offset_conv_kernel(
    const float* __restrict__ xin, const float* __restrict__ w_off,
    const float* __restrict__ b_off, float* __restrict__ off_out)
{
    int p  = blockIdx.x * 256 + threadIdx.x;
    int xw = p % W_;
    int t  = p / W_;
    int yh = t % H_;
    t /= H_;
    int o  = t % 18;
    int b  = t / 18;

    float acc = b_off[o];
    const float* xb = xin  + (size_t)b * CC * HW;
    const float* wo = w_off + (size_t)o * CC * 9;

    for (int ky = 0; ky < 3; ++ky) {
        int yy = yh + ky - 1;
        if (yy < 0 || yy >= H_) continue;
        for (int kx = 0; kx < 3; ++kx) {
            int xx = xw + kx - 1;
            if (xx < 0 || xx >= W_) continue;
            const float* xp = xb + yy * W_ + xx;
            const float* wp = wo + ky * 3 + kx;
            #pragma unroll 8
            for (int c = 0; c < CC; ++c)
                acc = fmaf(xp[c * HW], wp[c * 9], acc);
        }
    }
    off_out[p] = acc;
}

// ---------------------------------------------------------------------------
// Kernel 2: bilinear prep. For each (b, tap k, pixel): 4 clamped gather
// indices + 4 bilinear weights (validity folded into weight => zero padding,
// identical to the reference's masked gather).  B*KT*HW = 331776 threads.
// ---------------------------------------------------------------------------
__global__ __launch_bounds__(256) void bilinear_prep_kernel(
    const float* __restrict__ off, int4* __restrict__ idx_ws,
    float4* __restrict__ wgt_ws)
{
    int p  = blockIdx.x * 256 + threadIdx.x;
    int xw = p % W_;
    int t  = p / W_;
    int yh = t % H_;
    t /= H_;
    int k  = t % KT;
    int b  = t / KT;
    int pix = yh * W_ + xw;

    // offset channels laid out as [B][K][2][H][W] -> chan = 2k (dy), 2k+1 (dx)
    const float* ob = off + ((size_t)(b * 18 + 2 * k)) * HW + pix;
    float dy = ob[0];
    float dx = ob[HW];

    float py = (float)yh + (float)(k / 3 - 1) + dy;
    float px = (float)xw + (float)(k % 3 - 1) + dx;

    float y0f = floorf(py), x0f = floorf(px);
    float ly = py - y0f, lx = px - x0f;
    int y0 = (int)y0f, x0 = (int)x0f;
    int y1 = y0 + 1,   x1 = x0 + 1;
    float wy0 = 1.0f - ly, wx0 = 1.0f - lx;

    bool vy0 = (y0 >= 0) && (y0 < H_);
    bool vy1 = (y1 >= 0) && (y1 < H_);
    bool vx0 = (x0 >= 0) && (x0 < W_);
    bool vx1 = (x1 >= 0) && (x1 < W_);

    int y0c = min(max(y0, 0), H_ - 1), y1c = min(max(y1, 0), H_ - 1);
    int x0c = min(max(x0, 0), W_ - 1), x1c = min(max(x1, 0), W_ - 1);

    int4 i4;
    i4.x = y0c * W_ + x0c;   // v00
    i4.y = y0c * W_ + x1c;   // v01
    i4.z = y1c * W_ + x0c;   // v10
    i4.w = y1c * W_ + x1c;   // v11
    float4 w4;
    w4.x = (vy0 && vx0) ? wy0 * wx0 : 0.0f;
    w4.y = (vy0 && vx1) ? wy0 * lx  : 0.0f;
    w4.z = (vy1 && vx0) ? ly  * wx0 : 0.0f;
    w4.w = (vy1 && vx1) ? ly  * lx  : 0.0f;

    idx_ws[p] = i4;
    wgt_ws[p] = w4;
}

// ---------------------------------------------------------------------------
// Kernel 3: WMMA GEMM.  D[64 x 16] tile per wave, K-dim = 576, fp32 WMMA.
//   A = w_dc reshaped [64][576] (q = c*9 + k),  B[q][n] = bilinear sample.
// 2304 waves total (4 batches x 576 n-tiles), 8 waves per block, no
// divergent control flow (EXEC all ones as WMMA requires).
// ---------------------------------------------------------------------------
__global__ __launch_bounds__(256) void deform_gemm_kernel(
    const float* __restrict__ xin, const float* __restrict__ w_dc,
    const float* __restrict__ b_dc, const int4* __restrict__ idx_ws,
    const float4* __restrict__ wgt_ws, float* __restrict__ out)
{
    int wave = blockIdx.x * 8 + (threadIdx.x >> 5);
    int l  = threadIdx.x & 31;
    int ln = l & 15;             // column / row-in-fragment
    int lk = (l >> 4) << 1;      // k sub-offset within chunk of 4 {0,2}

    int b  = wave / (HW / 16);
    int nt = wave % (HW / 16);
    int n  = nt * 16 + ln;       // pixel index handled by this lane (B frag)

    const float*  xb   = xin    + (size_t)b * CC * HW;
    const int4*   idxb = idx_ws + (size_t)b * KT * HW;
    const float4* wgtb = wgt_ws + (size_t)b * KT * HW;

    v8f acc[4];
    acc[0] = v8f{}; acc[1] = v8f{}; acc[2] = v8f{}; acc[3] = v8f{};

    for (int q0 = 0; q0 < QDIM; q0 += 4) {
        // ---- build B fragment: 2 bilinear-sampled values per lane ----
        v2f bfrag;
        #pragma unroll
        for (int j = 0; j < 2; ++j) {
            int q = q0 + lk + j;       // q = c*9 + k
            int c = q / 9;
            int k = q - c * 9;
            int4   i4 = idxb[k * HW + n];
            float4 w4 = wgtb[k * HW + n];
            const float* xc = xb + c * HW;
            bfrag[j] = w4.x * xc[i4.x] + w4.y * xc[i4.y]
                     + w4.z * xc[i4.z] + w4.w * xc[i4.w];
        }
        // ---- 4 M-tiles share the gathered B fragment ----
        int qa = q0 + lk;
        #pragma unroll
        for (int mt = 0; mt < 4; ++mt) {
            const float* wrow = w_dc + (size_t)(mt * 16 + ln) * QDIM + qa;
            v2f afrag;
            afrag[0] = wrow[0];
            afrag[1] = wrow[1];
            acc[mt] = __builtin_amdgcn_wmma_f32_16x16x4_f32(
                false, afrag, false, bfrag, (short)0, acc[mt], false, false);
        }
    }

    // ---- epilogue: D layout  M = v + 8*(lane>>4),  N = lane&15 ----
    int mo = (l >> 4) * 8;
    #pragma unroll
    for (int mt = 0; mt < 4; ++mt) {
        #pragma unroll
        for (int v = 0; v < 8; ++v) {
            int m = mt * 16 + mo + v;
            out[((size_t)(b * CC + m)) * HW + nt * 16 + ln] = acc[mt][v] + b_dc[m];
        }
    }
}

// ---------------------------------------------------------------------------
extern "C" void kernel_launch(void* const* d_in, const int* in_sizes, int n_in,
                              void* d_out, int out_size, void* d_ws, size_t ws_size,
                              hipStream_t stream) {
    const float* x     = (const float*)d_in[0];   // [4,64,96,96]
    const float* w_off = (const float*)d_in[1];   // [18,64,3,3]
    const float* b_off = (const float*)d_in[2];   // [18]
    const float* w_dc  = (const float*)d_in[3];   // [64,64,3,3]
    const float* b_dc  = (const float*)d_in[4];   // [64]
    float* out = (float*)d_out;                   // [4,64,96,96]

    // workspace layout (16B aligned regions)
    const size_t OFF_ELEMS = (size_t)BB * 18 * HW;   // 663552 floats
    const size_t BKP_ELEMS = (size_t)BB * KT * HW;   // 331776 int4 / float4
    float*  off_ws = (float*)d_ws;
    int4*   idx_ws = (int4*)((char*)d_ws + OFF_ELEMS * sizeof(float));
    float4* wgt_ws = (float4*)((char*)d_ws + OFF_ELEMS * sizeof(float)
                                            + BKP_ELEMS * sizeof(int4));
    (void)ws_size; (void)in_sizes; (void)n_in; (void)out_size;

    // 1) offset conv: B*18*HW = 663552 threads
    offset_conv_kernel<<<663552 / 256, 256, 0, stream>>>(x, w_off, b_off, off_ws);
    // 2) bilinear index/weight prep: B*9*HW = 331776 threads
    bilinear_prep_kernel<<<331776 / 256, 256, 0, stream>>>(off_ws, idx_ws, wgt_ws);
    // 3) WMMA GEMM: 2304 waves, 8 waves (256 threads) per block
    deform_gemm_kernel<<<2304 / 8, 256, 0, stream>>>(x, w_dc, b_dc, idx_ws, wgt_ws, out);
}